// GIN_88656714925434
// MI455X (gfx1250) — compile-verified
//
#include <hip/hip_runtime.h>
#include <hip/hip_bf16.h>

// ---------------------------------------------------------------------------
// GIN (3x GINConv D=128 + mean-pool + FC 128->64) for MI455X / gfx1250.
// Edge aggregation is HBM-bound (~5 GB traffic): wave-per-edge float4 gather
// + f32 atomic scatter. Node GEMMs use V_WMMA_F32_16X16X4_F32 (fp32 WMMA).
// ---------------------------------------------------------------------------

typedef __attribute__((ext_vector_type(2))) float v2f;
typedef __attribute__((ext_vector_type(8))) float v8f;

#define N_NODES 100000
#define N_GRAPHS 512
#define DCH 128
#define LDS_PITCH 132   // 128 + 4 pad: avoids 16-way LDS bank conflicts on A reads

// --- Kernel 1: agg[dst[e]] += h[src[e]] --------------------------------------
// One wave32 per edge; lane q moves channel quad q (4 floats), 32 quads = 128.
__global__ void gin_scatter_kernel(const float* __restrict__ h,
                                   const int* __restrict__ src,
                                   const int* __restrict__ dst,
                                   float* __restrict__ agg, int E) {
    long long tid = (long long)blockIdx.x * blockDim.x + threadIdx.x;
    int e = (int)(tid >> 5);
    int q = (int)(tid & 31);
    if (e >= E) return;
    int s = src[e];              // wave-uniform load
    int d = dst[e];              // wave-uniform load
    const float4 v = reinterpret_cast<const float4*>(h + (size_t)s * DCH)[q];
    float* base = agg + (size_t)d * DCH + q * 4;
    atomicAdd(base + 0, v.x);
    atomicAdd(base + 1, v.y);
    atomicAdd(base + 2, v.z);
    atomicAdd(base + 3, v.w);
}

// --- Kernel 2: Y = relu((X + AGG) @ W + bias), X:[rows,128] W:[128,128] ------
// Block = 256 threads = 8 waves. Block owns 16 rows; wave w owns cols [16w,16w+16).
// Accumulates 32 x v_wmma_f32_16x16x4_f32 over K=128.
__global__ void __launch_bounds__(256)
gin_gemm_relu_kernel(const float* __restrict__ X,
                     const float* __restrict__ AGG,
                     const float* __restrict__ W,
                     const float* __restrict__ bias,
                     float* __restrict__ Y) {
    __shared__ float atile[16 * LDS_PITCH];

    const int t    = threadIdx.x;
    const int row0 = blockIdx.x * 16;

    // Stage A = X + AGG (16x128) into LDS, coalesced: 2048 floats / 256 threads.
    #pragma unroll
    for (int i = 0; i < 8; ++i) {
        int f   = t + i * 256;            // 0..2047
        int r   = f >> 7;                 // row 0..15
        int c   = f & 127;                // col 0..127
        size_t g = (size_t)(row0 + r) * DCH + c;
        atile[r * LDS_PITCH + c] = X[g] + AGG[g];
    }
    __syncthreads();

    const int lane = t & 31;
    const int wv   = t >> 5;              // 0..7 -> column tile
    const int n0   = wv * 16;
    const int m    = lane & 15;           // also n for B/C fragments
    const int khi  = (lane >> 4) * 2;     // lanes 16-31 hold K+2,K+3

    v8f c = {};
    #pragma unroll
    for (int kk = 0; kk < 32; ++kk) {
        const int k = kk * 4;
        v2f a, b;
        // A 16x4 fragment (ISA layout): lane<16 -> K={k,k+1}, lane>=16 -> K={k+2,k+3}
        a.x = atile[m * LDS_PITCH + k + khi];
        a.y = atile[m * LDS_PITCH + k + khi + 1];
        // B 4x16 fragment: row K striped across lanes, same K split as A
        b.x = W[(size_t)(k + khi) * DCH + n0 + m];
        b.y = W[(size_t)(k + khi + 1) * DCH + n0 + m];
        c = __builtin_amdgcn_wmma_f32_16x16x4_f32(
                /*neg_a=*/false, a, /*neg_b=*/false, b,
                /*c_mod=*/(short)0, c, /*reuse_a=*/false, /*reuse_b=*/false);
    }

    const float bb = bias[n0 + m];
    // C/D layout: VGPR v -> M=v (lanes 0-15) / M=v+8 (lanes 16-31), N=lane&15
    #pragma unroll
    for (int v = 0; v < 8; ++v) {
        int mr = v + (lane >> 4) * 8;
        float val = c[v] + bb;
        val = val > 0.0f ? val : 0.0f;
        Y[(size_t)(row0 + mr) * DCH + n0 + m] = val;
    }
}

// --- Kernel 3: graph pooling sums + counts -----------------------------------
__global__ void gin_pool_kernel(const float* __restrict__ h,
                                const int* __restrict__ batch,
                                float* __restrict__ sums,
                                float* __restrict__ counts) {
    long long tid = (long long)blockIdx.x * blockDim.x + threadIdx.x;
    int node = (int)(tid >> 5);
    int q    = (int)(tid & 31);
    if (node >= N_NODES) return;
    int g = batch[node];              // wave-uniform
    const float4 v = reinterpret_cast<const float4*>(h + (size_t)node * DCH)[q];
    float* base = sums + (size_t)g * DCH + q * 4;
    atomicAdd(base + 0, v.x);
    atomicAdd(base + 1, v.y);
    atomicAdd(base + 2, v.z);
    atomicAdd(base + 3, v.w);
    if (q == 0) atomicAdd(&counts[g], 1.0f);
}

// --- Kernel 4: out = (sums/count) @ Wfc + bfc  ([512,128]@[128,64]) ----------
__global__ void gin_fc_kernel(const float* __restrict__ sums,
                              const float* __restrict__ counts,
                              const float* __restrict__ Wfc,
                              const float* __restrict__ bfc,
                              float* __restrict__ out) {
    int idx = blockIdx.x * blockDim.x + threadIdx.x;
    if (idx >= N_GRAPHS * 64) return;
    int g = idx >> 6;
    int o = idx & 63;
    float cnt = counts[g];
    float inv = 1.0f / (cnt > 1.0f ? cnt : 1.0f);
    float acc = bfc[o];
    const float* sg = sums + (size_t)g * DCH;
    #pragma unroll 4
    for (int k = 0; k < DCH; ++k)
        acc += (sg[k] * inv) * Wfc[(size_t)k * 64 + o];
    out[(size_t)g * 64 + o] = acc;
}

// ---------------------------------------------------------------------------
extern "C" void kernel_launch(void* const* d_in, const int* in_sizes, int n_in,
                              void* d_out, int out_size, void* d_ws, size_t ws_size,
                              hipStream_t stream) {
    const float* x    = (const float*)d_in[0];
    const int*  eidx  = (const int*)d_in[1];
    const int*  batch = (const int*)d_in[2];
    const float* W1   = (const float*)d_in[3];
    const float* b1   = (const float*)d_in[4];
    const float* W2   = (const float*)d_in[5];
    const float* b2   = (const float*)d_in[6];
    const float* W3   = (const float*)d_in[7];
    const float* b3   = (const float*)d_in[8];
    const float* Wfc  = (const float*)d_in[9];
    const float* bfc  = (const float*)d_in[10];
    float* out = (float*)d_out;

    const int E = in_sizes[1] / 2;
    const int* src = eidx;
    const int* dst = eidx + E;

    // Workspace layout
    const size_t nodeBytes = (size_t)N_NODES * DCH * sizeof(float);   // 51.2 MB
    char* ws   = (char*)d_ws;
    float* agg = (float*)(ws);
    float* h1  = (float*)(ws + nodeBytes);
    float* h2  = (float*)(ws + 2 * nodeBytes);
    float* sums   = (float*)(ws + 3 * nodeBytes);
    float* counts = (float*)(ws + 3 * nodeBytes + (size_t)N_GRAPHS * DCH * sizeof(float));

    const long long scatterThreads = (long long)E * 32;
    const int scatterBlocks = (int)((scatterThreads + 255) / 256);
    const int gemmBlocks = N_NODES / 16;   // 6250, exact
    const long long poolThreads = (long long)N_NODES * 32;
    const int poolBlocks = (int)((poolThreads + 255) / 256);

    // Layer 1
    hipMemsetAsync(agg, 0, nodeBytes, stream);
    gin_scatter_kernel<<<scatterBlocks, 256, 0, stream>>>(x, src, dst, agg, E);
    gin_gemm_relu_kernel<<<gemmBlocks, 256, 0, stream>>>(x, agg, W1, b1, h1);
    // Layer 2
    hipMemsetAsync(agg, 0, nodeBytes, stream);
    gin_scatter_kernel<<<scatterBlocks, 256, 0, stream>>>(h1, src, dst, agg, E);
    gin_gemm_relu_kernel<<<gemmBlocks, 256, 0, stream>>>(h1, agg, W2, b2, h2);
    // Layer 3
    hipMemsetAsync(agg, 0, nodeBytes, stream);
    gin_scatter_kernel<<<scatterBlocks, 256, 0, stream>>>(h2, src, dst, agg, E);
    gin_gemm_relu_kernel<<<gemmBlocks, 256, 0, stream>>>(h2, agg, W3, b3, h1);
    // Pool + FC
    hipMemsetAsync(sums, 0, (size_t)N_GRAPHS * DCH * sizeof(float) +
                            (size_t)N_GRAPHS * sizeof(float), stream);
    gin_pool_kernel<<<poolBlocks, 256, 0, stream>>>(h1, batch, sums, counts);
    gin_fc_kernel<<<(N_GRAPHS * 64 + 255) / 256, 256, 0, stream>>>(
        sums, counts, Wfc, bfc, out);
}